// SchInteraction_44590350467107
// MI455X (gfx1250) — compile-verified
//
#include <hip/hip_runtime.h>
#include <math.h>

#define DFEAT 64
#define WP_STRIDE 96    // floats per k-pair row: shift between lane-halves = 96 mod 64 = 32 banks
                        // -> lanes 0-15 hit banks 0-31, lanes 16-31 hit banks 32-63: conflict-free b64
#define XT_STRIDE 68    // tile LDS row stride: conflict-free b64 A-fragment reads, 16B-aligned float4 stores
#define NWAVES 8
#define TPB (NWAVES * 32)

typedef __attribute__((ext_vector_type(2))) float v2f;
typedef __attribute__((ext_vector_type(8))) float v8f;

__device__ __forceinline__ float sspf(float x) {
    // shifted softplus: log(1+exp(x)) - log(2), numerically stable
    return fmaxf(x, 0.0f) + log1pf(__expf(-fabsf(x))) - 0.69314718055994531f;
}

// Stage a 64x64 row-major weight matrix into LDS in pair-major layout:
//   wl[(k>>1)*WP_STRIDE + n*2 + (k&1)] = W[k*64 + n]
// so the WMMA B-fragment (W[k0][n], W[k0+1][n]) is one contiguous, 8B-aligned b64.
__device__ __forceinline__ void load_weights_lds(float* __restrict__ wlds,
                                                 const float* __restrict__ W, int tid) {
#pragma unroll
    for (int i = 0; i < (DFEAT * DFEAT) / TPB; ++i) {
        int idx = tid + i * TPB;
        int k = idx >> 6, n = idx & 63;
        wlds[(k >> 1) * WP_STRIDE + n * 2 + (k & 1)] = W[idx];
    }
}

// Stage a [16,64] fp32 tile from global into this wave's LDS tile (float4 coalesced).
__device__ __forceinline__ void load_tile_lds(float* __restrict__ xt,
                                              const float* __restrict__ g,
                                              long row0, int lane) {
#pragma unroll
    for (int i = 0; i < 8; ++i) {
        int idx = lane * 4 + i * 128;          // 0..1023
        int row = idx >> 6, col = idx & 63;
        float4 v = *(const float4*)(g + (row0 + row) * DFEAT + col);
        *(float4*)(xt + row * XT_STRIDE + col) = v;   // (row*68+col)*4 is 16B aligned
    }
}

// One wave computes Y[16,64] += X[16,64] @ W[64,64] via 64x v_wmma_f32_16x16x4_f32.
// A layout (16x4 f32): lane m=lane&15, khalf=lane>>4; a[i] = X[m][4*kt + 2*khalf + i]
// B layout (4x16 f32): lane n=lane&15, khalf=lane>>4; b[i] = W[4*kt + 2*khalf + i][nt*16+n]
//   = wl[(2*kt+khalf)*WP_STRIDE + (nt*16+n)*2 + i]  -> single ds_load_b64 per fragment
__device__ __forceinline__ void wave_gemm(const float* __restrict__ xt,
                                          const float* __restrict__ wt,
                                          v8f acc[4], int lane) {
    const int m = lane & 15;
    const int khalf = lane >> 4;
    const int n2 = (lane & 15) * 2;
#pragma unroll
    for (int kt = 0; kt < 16; ++kt) {
        v2f a = *(const v2f*)(xt + m * XT_STRIDE + kt * 4 + khalf * 2);
        const float* wrow = wt + (2 * kt + khalf) * WP_STRIDE + n2;
#pragma unroll
        for (int nt = 0; nt < 4; ++nt) {
            v2f b = *(const v2f*)(wrow + nt * 32);
            acc[nt] = __builtin_amdgcn_wmma_f32_16x16x4_f32(
                false, a, false, b, (short)0, acc[nt], false, false);
        }
    }
}

// Write C/D-layout accumulators back into the wave's LDS tile (row-major) for chaining.
__device__ __forceinline__ void stage_acc_lds(float* __restrict__ xt,
                                              const v8f acc[4], int lane) {
    const int n = lane & 15;
    const int lh = lane >> 4;
#pragma unroll
    for (int nt = 0; nt < 4; ++nt)
#pragma unroll
        for (int r = 0; r < 8; ++r)
            xt[(r + 8 * lh) * XT_STRIDE + nt * 16 + n] = acc[nt][r];
}

// ---------------- kernel 1: h = node_feats @ Wn + bn ----------------
__global__ __launch_bounds__(TPB) void schnet_node_proj(
        const float* __restrict__ node, const float* __restrict__ Wn,
        const float* __restrict__ bn, float* __restrict__ h, int numTiles) {
    extern __shared__ float smem[];
    float* wlds = smem;
    const int tid = threadIdx.x, lane = tid & 31, wave = tid >> 5;
    float* xt = smem + 32 * WP_STRIDE + wave * (16 * XT_STRIDE);
    load_weights_lds(wlds, Wn, tid);
    __syncthreads();
    const int n = lane & 15, lh = lane >> 4;
    float bv[4];
#pragma unroll
    for (int nt = 0; nt < 4; ++nt) bv[nt] = bn[nt * 16 + n];
    for (int t = blockIdx.x * NWAVES + wave; t < numTiles; t += gridDim.x * NWAVES) {
        long row0 = (long)t * 16;
        load_tile_lds(xt, node, row0, lane);
        v8f acc[4] = {};
        wave_gemm(xt, wlds, acc, lane);
#pragma unroll
        for (int nt = 0; nt < 4; ++nt)
#pragma unroll
            for (int r = 0; r < 8; ++r)
                h[(row0 + r + 8 * lh) * DFEAT + nt * 16 + n] = acc[nt][r] + bv[nt];
    }
}

// -------- kernel 2: fused edge MLP + gather/modulate/scatter-add --------
__global__ __launch_bounds__(TPB) void schnet_edge(
        const float* __restrict__ ef,
        const float* __restrict__ We1, const float* __restrict__ be1,
        const float* __restrict__ We2, const float* __restrict__ be2,
        const int* __restrict__ src, const int* __restrict__ dst,
        const float* __restrict__ h, float* __restrict__ agg, int numTiles) {
    extern __shared__ float smem[];
    float* w1 = smem;
    float* w2 = smem + 32 * WP_STRIDE;
    const int tid = threadIdx.x, lane = tid & 31, wave = tid >> 5;
    float* xt = smem + 64 * WP_STRIDE + wave * (16 * XT_STRIDE);
    load_weights_lds(w1, We1, tid);
    load_weights_lds(w2, We2, tid);
    __syncthreads();
    const int n = lane & 15, lh = lane >> 4;
    float b1v[4], b2v[4];
#pragma unroll
    for (int nt = 0; nt < 4; ++nt) {
        b1v[nt] = be1[nt * 16 + n];
        b2v[nt] = be2[nt * 16 + n];
    }
    for (int t = blockIdx.x * NWAVES + wave; t < numTiles; t += gridDim.x * NWAVES) {
        long e0 = (long)t * 16;
        load_tile_lds(xt, ef, e0, lane);
        v8f acc[4] = {};
        wave_gemm(xt, w1, acc, lane);                 // layer 1
#pragma unroll
        for (int nt = 0; nt < 4; ++nt)
#pragma unroll
            for (int r = 0; r < 8; ++r)
                acc[nt][r] = sspf(acc[nt][r] + b1v[nt]);
        stage_acc_lds(xt, acc, lane);                 // round-trip C/D -> A layout
        v8f fa[4] = {};
        wave_gemm(xt, w2, fa, lane);                  // layer 2
#pragma unroll
        for (int nt = 0; nt < 4; ++nt)
#pragma unroll
            for (int r = 0; r < 8; ++r)
                fa[nt][r] = sspf(fa[nt][r] + b2v[nt]);
        // gather h[src], modulate, scatter-add to agg[dst] (L2-resident, coalesced per row)
#pragma unroll
        for (int r = 0; r < 8; ++r) {
            long e = e0 + r + 8 * lh;
            int s = src[e], d = dst[e];
#pragma unroll
            for (int nt = 0; nt < 4; ++nt) {
                float msg = h[(long)s * DFEAT + nt * 16 + n] * fa[nt][r];
                atomicAdd(&agg[(long)d * DFEAT + nt * 16 + n], msg);
            }
        }
    }
}

// ------ kernel 3: out = ssp(agg @ Wc + bc) @ Wo + bo ------
__global__ __launch_bounds__(TPB) void schnet_out(
        const float* __restrict__ agg,
        const float* __restrict__ Wc, const float* __restrict__ bc,
        const float* __restrict__ Wo, const float* __restrict__ bo,
        float* __restrict__ out, int numTiles) {
    extern __shared__ float smem[];
    float* w1 = smem;
    float* w2 = smem + 32 * WP_STRIDE;
    const int tid = threadIdx.x, lane = tid & 31, wave = tid >> 5;
    float* xt = smem + 64 * WP_STRIDE + wave * (16 * XT_STRIDE);
    load_weights_lds(w1, Wc, tid);
    load_weights_lds(w2, Wo, tid);
    __syncthreads();
    const int n = lane & 15, lh = lane >> 4;
    float b1v[4], b2v[4];
#pragma unroll
    for (int nt = 0; nt < 4; ++nt) {
        b1v[nt] = bc[nt * 16 + n];
        b2v[nt] = bo[nt * 16 + n];
    }
    for (int t = blockIdx.x * NWAVES + wave; t < numTiles; t += gridDim.x * NWAVES) {
        long row0 = (long)t * 16;
        load_tile_lds(xt, agg, row0, lane);
        v8f acc[4] = {};
        wave_gemm(xt, w1, acc, lane);
#pragma unroll
        for (int nt = 0; nt < 4; ++nt)
#pragma unroll
            for (int r = 0; r < 8; ++r)
                acc[nt][r] = sspf(acc[nt][r] + b1v[nt]);
        stage_acc_lds(xt, acc, lane);
        v8f o[4] = {};
        wave_gemm(xt, w2, o, lane);
#pragma unroll
        for (int nt = 0; nt < 4; ++nt)
#pragma unroll
            for (int r = 0; r < 8; ++r)
                out[(row0 + r + 8 * lh) * DFEAT + nt * 16 + n] = o[nt][r] + b2v[nt];
    }
}

extern "C" void kernel_launch(void* const* d_in, const int* in_sizes, int n_in,
                              void* d_out, int out_size, void* d_ws, size_t ws_size,
                              hipStream_t stream) {
    const float* node = (const float*)d_in[0];
    const float* ef   = (const float*)d_in[1];
    const int*   src  = (const int*)d_in[2];
    const int*   dst  = (const int*)d_in[3];
    const float* We1  = (const float*)d_in[4];
    const float* be1  = (const float*)d_in[5];
    const float* We2  = (const float*)d_in[6];
    const float* be2  = (const float*)d_in[7];
    const float* Wn   = (const float*)d_in[8];
    const float* bn   = (const float*)d_in[9];
    const float* Wc   = (const float*)d_in[10];
    const float* bc   = (const float*)d_in[11];
    const float* Wo   = (const float*)d_in[12];
    const float* bo   = (const float*)d_in[13];
    float* out = (float*)d_out;

    const int  V = in_sizes[0] / DFEAT;   // 50000
    const long E = in_sizes[2];           // 1250000

    float* h   = (float*)d_ws;
    float* agg = h + (size_t)V * DFEAT;

    hipMemsetAsync(agg, 0, (size_t)V * DFEAT * sizeof(float), stream);

    const int nodeTiles = V / 16;             // 3125 (exact)
    const int edgeTiles = (int)(E / 16);      // 78125 (exact)

    const size_t shm1 = (size_t)(32 * WP_STRIDE + NWAVES * 16 * XT_STRIDE) * sizeof(float);
    const size_t shm2 = (size_t)(64 * WP_STRIDE + NWAVES * 16 * XT_STRIDE) * sizeof(float);

    int gridN = (nodeTiles + NWAVES - 1) / NWAVES;     // 391
    int gridE = 1024;
    if ((long)gridE * NWAVES > edgeTiles) gridE = (edgeTiles + NWAVES - 1) / NWAVES;

    schnet_node_proj<<<gridN, TPB, shm1, stream>>>(node, Wn, bn, h, nodeTiles);
    schnet_edge<<<gridE, TPB, shm2, stream>>>(ef, We1, be1, We2, be2, src, dst, h, agg, edgeTiles);
    schnet_out<<<gridN, TPB, shm2, stream>>>(agg, Wc, bc, Wo, bo, out, nodeTiles);
}